// EvalMemoryReader_32770600468514
// MI455X (gfx1250) — compile-verified
//
#include <hip/hip_runtime.h>
#include <hip/hip_bf16.h>

// Problem constants (from reference: B=1, CK=64, CV=512, T=12, H=36, W=64)
#define CK    64
#define CV    512
#define W_    64
#define HW    2304      // 36*64
#define THW   27648     // 12*2304
#define TOPK  50
#define INV2S2 0.010204081632653061f   // 1/(2*7^2)
#define QSCALE 0.125f                  // 1/sqrt(64)

typedef __attribute__((ext_vector_type(2))) float v2f;
typedef __attribute__((ext_vector_type(8))) float v8f;
typedef __attribute__((ext_vector_type(4))) int   v4i;

#define AS1 __attribute__((address_space(1)))
#define AS3 __attribute__((address_space(3)))

#if __has_builtin(__builtin_amdgcn_global_load_async_to_lds_b128) && \
    __has_builtin(__builtin_amdgcn_s_wait_asynccnt)
#define HAVE_ASYNC_LDS 1
#else
#define HAVE_ASYNC_LDS 0
#endif

// ---------------------------------------------------------------------------
// Kernel A: affinity[t,h] = sum_c (mk[c,t]*QSCALE) * qk[c,h]
// GEMM M=27648 N=2304 K=64 via V_WMMA_F32_16X16X4_F32 (fp32 exact).
// Each wave computes a 16(M) x 64(N) strip: the A fragment is reused across
// 4 WMMAs per k-step (4x better load/FLOP ratio than one tile per wave).
// ---------------------------------------------------------------------------
__global__ __launch_bounds__(256) void affinity_wmma(const float* __restrict__ mk,
                                                     const float* __restrict__ qk,
                                                     float* __restrict__ aff) {
  const int lane = threadIdx.x & 31;
  const int wave = threadIdx.x >> 5;
  const int n0 = blockIdx.x * 64;                   // h strip base (4 tiles)
  const int m0 = (blockIdx.y * 8 + wave) * 16;      // t tile base
  const int r16  = lane & 15;                       // m (for A) / n (for B)
  const int half = lane >> 4;                       // 0: K={0,1}, 1: K={2,3}
  const int klo  = half * 2;

  v8f c0 = {}, c1 = {}, c2 = {}, c3 = {};
  #pragma unroll
  for (int k0 = 0; k0 < CK; k0 += 4) {
    // A = (mi^T) * QSCALE : A[m][k] = mk[k*THW + (m0+m)] * QSCALE
    v2f a;
    a.x = mk[(size_t)(k0 + klo    ) * THW + (m0 + r16)] * QSCALE;
    a.y = mk[(size_t)(k0 + klo + 1) * THW + (m0 + r16)] * QSCALE;
    // B = qi : B[k][n] = qk[k*HW + n]
    v2f b0, b1, b2, b3;
    const size_t brow0 = (size_t)(k0 + klo) * HW + n0 + r16;
    b0.x = qk[brow0     ]; b0.y = qk[brow0 + HW     ];
    b1.x = qk[brow0 + 16]; b1.y = qk[brow0 + HW + 16];
    b2.x = qk[brow0 + 32]; b2.y = qk[brow0 + HW + 32];
    b3.x = qk[brow0 + 48]; b3.y = qk[brow0 + HW + 48];
    // 8 args: (neg_a, A, neg_b, B, c_mod, C, reuse_a, reuse_b)
    c0 = __builtin_amdgcn_wmma_f32_16x16x4_f32(false, a, false, b0, (short)0, c0, false, false);
    c1 = __builtin_amdgcn_wmma_f32_16x16x4_f32(false, a, false, b1, (short)0, c1, false, false);
    c2 = __builtin_amdgcn_wmma_f32_16x16x4_f32(false, a, false, b2, (short)0, c2, false, false);
    c3 = __builtin_amdgcn_wmma_f32_16x16x4_f32(false, a, false, b3, (short)0, c3, false, false);
  }
  // C/D layout: VGPR v, lanes 0-15 -> M=v, lanes 16-31 -> M=v+8; N = lane%16
  const int rbase = m0 + 8 * half;
  #pragma unroll
  for (int v = 0; v < 8; ++v) {
    const size_t r = (size_t)(rbase + v) * HW + n0 + r16;
    aff[r     ] = c0[v];
    aff[r + 16] = c1[v];
    aff[r + 32] = c2[v];
    aff[r + 48] = c3[v];
  }
}

// ---------------------------------------------------------------------------
// Kernel B: per-row argmax over h (first-max tie-break like jnp.argmax)
// ---------------------------------------------------------------------------
__global__ __launch_bounds__(256) void row_argmax(const float* __restrict__ aff,
                                                  int* __restrict__ amax) {
  const int t = blockIdx.x;
  const float* row = aff + (size_t)t * HW;
  float best = -3.4e38f;
  int bi = 0;
  for (int h = threadIdx.x; h < HW; h += 256) {
    float v = row[h];
    if (v > best) { best = v; bi = h; }           // indices ascending per thread
  }
  __shared__ float sv[256];
  __shared__ int   si[256];
  sv[threadIdx.x] = best; si[threadIdx.x] = bi;
  __syncthreads();
  for (int s = 128; s > 0; s >>= 1) {
    if (threadIdx.x < s) {
      float ov = sv[threadIdx.x + s]; int oi = si[threadIdx.x + s];
      if (ov > sv[threadIdx.x] || (ov == sv[threadIdx.x] && oi < si[threadIdx.x])) {
        sv[threadIdx.x] = ov; si[threadIdx.x] = oi;
      }
    }
    __syncthreads();
  }
  if (threadIdx.x == 0) amax[t] = si[0];
}

// ---------------------------------------------------------------------------
// Kernel C: per-column max over t, via monotonic-uint atomicMax
// ---------------------------------------------------------------------------
__device__ __forceinline__ unsigned f2key(float f) {
  unsigned b = __float_as_uint(f);
  return (b & 0x80000000u) ? ~b : (b | 0x80000000u);
}
__device__ __forceinline__ float key2f(unsigned k) {
  return (k & 0x80000000u) ? __uint_as_float(k ^ 0x80000000u)
                           : __uint_as_float(~k);
}

__global__ __launch_bounds__(256) void init_maxkey(unsigned* __restrict__ mkey) {
  int h = blockIdx.x * 256 + threadIdx.x;
  if (h < HW) mkey[h] = 0u;
}

__global__ __launch_bounds__(256) void col_max(const float* __restrict__ aff,
                                               unsigned* __restrict__ mkey) {
  const int h  = blockIdx.x * 256 + threadIdx.x;   // coalesced across h
  const int r0 = blockIdx.y * 1024;
  float m = -3.4e38f;
  for (int r = r0; r < r0 + 1024; ++r)
    m = fmaxf(m, aff[(size_t)r * HW + h]);
  atomicMax(&mkey[h], f2key(m));
}

// ---------------------------------------------------------------------------
// Kernel D: per-column top-50 via 4x8-bit radix histogram threshold select.
// score = exp(aff - colmax - dist2/(2*sigma^2))  (gaussian folded into exp).
// score >= 0 so float bits are order-preserving as unsigned.
// ---------------------------------------------------------------------------
__device__ __forceinline__ unsigned score_key(const float* aff, const int* amax,
                                              int t, int h, float colmax,
                                              float yq, float xq) {
  float a  = aff[(size_t)t * HW + h];
  int   am = amax[t];
  float dy = yq - (float)(am / W_);
  float dx = xq - (float)(am % W_);
  float s  = a - colmax - (dy * dy + dx * dx) * INV2S2;
  return __float_as_uint(__expf(s));
}

__global__ __launch_bounds__(256) void topk_select(const float* __restrict__ aff,
                                                   const int* __restrict__ amax,
                                                   const unsigned* __restrict__ mkey,
                                                   int* __restrict__ tki,
                                                   float* __restrict__ tkw) {
  const int h = blockIdx.x;
  const float colmax = key2f(mkey[h]);
  const float yq = (float)(h / W_), xq = (float)(h % W_);

  __shared__ unsigned hist[256];
  __shared__ unsigned sh_prefix, sh_above;
  if (threadIdx.x == 0) { sh_prefix = 0u; sh_above = 0u; }
  __syncthreads();

  // 4 radix passes: resolve the 50th-largest key exactly (8 bits per pass)
  for (int p = 3; p >= 0; --p) {
    hist[threadIdx.x] = 0u;
    __syncthreads();
    const unsigned prefix = sh_prefix;
    const unsigned pmask  = (p == 3) ? 0u : (0xFFFFFFFFu << ((p + 1) * 8));
    for (int t = threadIdx.x; t < THW; t += 256) {
      unsigned key = score_key(aff, amax, t, h, colmax, yq, xq);
      if ((key & pmask) == prefix)
        atomicAdd(&hist[(key >> (p * 8)) & 255], 1u);
    }
    __syncthreads();
    if (threadIdx.x == 0) {
      unsigned cum = sh_above;
      int b = 255;
      for (; b > 0; --b) {
        if (cum + hist[b] >= (unsigned)TOPK) break;
        cum += hist[b];
      }
      sh_prefix = prefix | ((unsigned)b << (p * 8));
      sh_above  = cum;                    // count of keys strictly > new prefix
    }
    __syncthreads();
  }
  const unsigned thr   = sh_prefix;
  const unsigned above = sh_above;        // < TOPK
  const int need = TOPK - (int)above;     // ties to take at == thr

  __shared__ int   cg, ce;
  __shared__ int   s_idx[TOPK];
  __shared__ float s_val[TOPK];
  if (threadIdx.x == 0) { cg = 0; ce = 0; }
  if (threadIdx.x < TOPK) { s_idx[threadIdx.x] = 0; s_val[threadIdx.x] = 0.f; }
  __syncthreads();

  for (int t = threadIdx.x; t < THW; t += 256) {
    unsigned key = score_key(aff, amax, t, h, colmax, yq, xq);
    if (key > thr) {
      int s = atomicAdd(&cg, 1);
      if (s < (int)above) { s_idx[s] = t; s_val[s] = __uint_as_float(key); }
    } else if (key == thr) {
      int e = atomicAdd(&ce, 1);
      if (e < need) {
        int s = (int)above + e;
        s_idx[s] = t; s_val[s] = __uint_as_float(key);
      }
    }
  }
  __syncthreads();

  __shared__ float ssum;
  if (threadIdx.x == 0) {
    float s = 0.f;
    for (int i = 0; i < TOPK; ++i) s += s_val[i];
    ssum = s;
  }
  __syncthreads();
  float inv = (ssum > 0.f) ? (1.0f / ssum) : 0.f;
  if (threadIdx.x < TOPK) {
    tki[h * TOPK + threadIdx.x] = s_idx[threadIdx.x];
    tkw[h * TOPK + threadIdx.x] = s_val[threadIdx.x] * inv;
  }
}

// ---------------------------------------------------------------------------
// Kernel E: out[c,h] = sum_k mv[c, idx[h,k]] * w[h,k]
// Stage mv row (110KB) in LDS in two 55KB chunks -> all gathers hit LDS.
// Uses GLOBAL_LOAD_ASYNC_TO_LDS_B128 (ASYNCcnt) when the builtin is available.
// ---------------------------------------------------------------------------
#define ECHUNK 13824   // THW/2
__global__ __launch_bounds__(256) void sparse_mm(const float* __restrict__ mv,
                                                 const int* __restrict__ tki,
                                                 const float* __restrict__ tkw,
                                                 float* __restrict__ out) {
  const int c = blockIdx.x;                         // 0..CV-1
  __shared__ float row[ECHUNK];
  float acc[HW / 256] = {};
  const float* mrow = mv + (size_t)c * THW;

  for (int chunk = 0; chunk < 2; ++chunk) {
    const int base = chunk * ECHUNK;
    __syncthreads();
#if HAVE_ASYNC_LDS
    {
      // 16B per transfer, direct to LDS (no VGPR round-trip)
      for (int i = threadIdx.x * 4; i < ECHUNK; i += 256 * 4) {
        AS1 v4i* g = (AS1 v4i*)(uintptr_t)(mrow + base + i);
        AS3 v4i* l = (AS3 v4i*)(uint32_t)(uintptr_t)(&row[i]);
        __builtin_amdgcn_global_load_async_to_lds_b128(g, l, 0, 0);
      }
      __builtin_amdgcn_s_wait_asynccnt(0);
    }
#else
    for (int i = threadIdx.x; i < ECHUNK; i += 256) row[i] = mrow[base + i];
#endif
    __syncthreads();
    #pragma unroll
    for (int j = 0; j < HW / 256; ++j) {
      const int h = threadIdx.x + 256 * j;
      const int*   ip = tki + h * TOPK;
      const float* wp = tkw + h * TOPK;
      for (int k = 0; k < TOPK; ++k) {
        int idx = ip[k] - base;
        if ((unsigned)idx < (unsigned)ECHUNK) acc[j] += wp[k] * row[idx];
      }
    }
  }
  #pragma unroll
  for (int j = 0; j < HW / 256; ++j)
    out[(size_t)c * HW + threadIdx.x + 256 * j] = acc[j];
}

// ---------------------------------------------------------------------------
extern "C" void kernel_launch(void* const* d_in, const int* in_sizes, int n_in,
                              void* d_out, int out_size, void* d_ws, size_t ws_size,
                              hipStream_t stream) {
  const float* mk = (const float*)d_in[0];   // [64, 27648]
  const float* mv = (const float*)d_in[1];   // [512, 27648]
  const float* qk = (const float*)d_in[2];   // [64, 2304]
  float* out = (float*)d_out;                // [512, 2304]

  // Workspace layout (~256 MB): affinity + argmax + colmax keys + topk lists
  float*    aff   = (float*)d_ws;                         // THW*HW floats
  int*      amax  = (int*)(aff + (size_t)THW * HW);       // THW
  unsigned* mkey  = (unsigned*)(amax + THW);              // HW
  int*      tki   = (int*)(mkey + HW);                    // HW*TOPK
  float*    tkw   = (float*)(tki + (size_t)HW * TOPK);    // HW*TOPK

  // 1) affinity GEMM (WMMA fp32): waves cover 16x64 strips
  affinity_wmma<<<dim3(HW / 64, THW / (16 * 8)), 256, 0, stream>>>(mk, qk, aff);
  // 2) per-row argmax (gaussian centers)
  row_argmax<<<THW, 256, 0, stream>>>(aff, amax);
  // 3) per-column max
  init_maxkey<<<(HW + 255) / 256, 256, 0, stream>>>(mkey);
  col_max<<<dim3(HW / 256, THW / 1024), 256, 0, stream>>>(aff, mkey);
  // 4) per-column top-50 with normalized weights
  topk_select<<<HW, 256, 0, stream>>>(aff, amax, mkey, tki, tkw);
  // 5) sparse gather-GEMM into output
  sparse_mm<<<CV, 256, 0, stream>>>(mv, tki, tkw, out);
}